// Dense_Q_MulIn1Out_Conv1D_84542136255138
// MI455X (gfx1250) — compile-verified
//
#include <hip/hip_runtime.h>
#include <math.h>

typedef __attribute__((ext_vector_type(2))) float v2f;
typedef __attribute__((ext_vector_type(8))) float v8f;

#define QDIM     512      // 2^9 state dimension
#define SROW     516      // padded LDS row stride in floats (516 mod 64 = 4 -> conflict-free col reads)
#define NQ       9
#define NLAYERS  3
#define LOUT     4089
#define NBATCH   16
#define CIN      16
#define DPATCH   128      // K * C_IN
#define MROWS    16       // psi rows per (single-wave) workgroup
#define LTOT     4096
#define NTILE    4        // n-tiles processed concurrently (4 independent WMMA acc chains)

__global__ __launch_bounds__(32)
void qconv_wmma_kernel(const float* __restrict__ x,
                       const float* __restrict__ E,      // (512,512) row-major
                       const float* __restrict__ theta,  // (3,9)
                       float* __restrict__ out)          // (16*4089)
{
    __shared__ float psiA[MROWS * SROW];
    __shared__ float psiB[MROWS * SROW];

    const int lane = threadIdx.x;            // 0..31
    const int m0   = blockIdx.x * MROWS;     // global patch-row base (4089 blocks, exact)

    // ---------------- 1. gather patch values into psiA[r][0..127] ----------------
    for (int i = 0; i < (MROWS * DPATCH) / 32; ++i) {
        int t = i * 32 + lane;
        int r = t >> 7;                      // row in slab
        int d = t & 127;                     // feature index = c*8 + k
        int c = d >> 3;
        int k = d & 7;
        int g = m0 + r;
        int b = g / LOUT;
        int l = g - b * LOUT;
        psiA[r * SROW + d] = x[((size_t)(b * CIN + c)) * LTOT + (size_t)(l + k)];
    }
    // zero pad region d = 128..511
    for (int i = 0; i < (MROWS * (QDIM - DPATCH)) / 32; ++i) {
        int t = i * 32 + lane;
        int r = t / (QDIM - DPATCH);
        int d = DPATCH + (t - r * (QDIM - DPATCH));
        psiA[r * SROW + d] = 0.0f;
    }
    __syncthreads();

    // ---------------- 2. normalize rows: lane pair (L, L+16) splits row L&15 ----------------
    {
        int row = lane & 15;
        int h   = lane >> 4;
        float sum = 0.0f;
        for (int d = h * 64; d < h * 64 + 64; ++d) {
            float v = psiA[row * SROW + d];
            sum += v * v;
        }
        sum += __shfl_xor(sum, 16, 32);
        float rinv = rsqrtf(sum + 1e-12f);
        for (int d = h * 64; d < h * 64 + 64; ++d)
            psiA[row * SROW + d] *= rinv;
    }
    __syncthreads();

    // WMMA fragment lane mapping (ISA 7.12.2, 32-bit A / 32-bit C/D, wave32)
    const int mm    = lane & 15;             // A-matrix row / B-matrix column
    const int khalf = (lane >> 4) * 2;       // K offset: lanes 0-15 -> K{0,1}, 16-31 -> K{2,3}
    const int mbase = (lane >> 4) * 8;       // C/D: VGPR v holds rows {v, v+8}

    float* src = psiA;
    float* dst = psiB;

    for (int layer = 0; layer < NLAYERS; ++layer) {
        // ---------- RY butterfly layer, in place on src ----------
        for (int q = 0; q < NQ; ++q) {
            float half = theta[layer * NQ + q] * 0.5f;
            float cq = cosf(half);
            float sq = sinf(half);
            int   sh = 8 - q;                // log2(stride between pair elements)
            int   right = 1 << sh;
            for (int i = 0; i < (MROWS * (QDIM / 2)) / 32; ++i) {   // 128 iters
                int p  = i * 32 + lane;      // pair id in slab
                int r  = p >> 8;             // row
                int t  = p & 255;            // pair id within row
                int li = t >> sh;
                int ri = t & (right - 1);
                int i0 = (li << (sh + 1)) + ri;
                int i1 = i0 + right;
                float a = src[r * SROW + i0];
                float b = src[r * SROW + i1];
                src[r * SROW + i0] = cq * a - sq * b;
                src[r * SROW + i1] = sq * a + cq * b;
            }
            __syncthreads();
        }

        // ---------- dst = src @ E^T via V_WMMA_F32_16X16X4_F32 ----------
        // out[m][n] = sum_k src[m][k] * E[n][k]
        // 4 n-tiles concurrently: 1 shared A fragment, 4 B fragments,
        // 4 independent accumulator chains -> 4-way ILP on the matrix pipe.
        const float* ab = src + mm * SROW + khalf;                  // LDS, b64 per lane
        for (int nt = 0; nt < QDIM / 16; nt += NTILE) {
            int n0 = nt * 16;
            const float* eb = E + (size_t)(n0 + mm) * QDIM + khalf; // global, b64 per lane
            if (nt + NTILE < QDIM / 16)
                __builtin_prefetch(eb + NTILE * 16 * QDIM, 0, 1);   // prefetch next tile group
            v8f acc0 = {}, acc1 = {}, acc2 = {}, acc3 = {};
            #pragma unroll 4
            for (int k0 = 0; k0 < QDIM; k0 += 4) {
                v2f a  = *(const v2f*)(ab + k0);
                v2f b0 = *(const v2f*)(eb + k0);
                v2f b1 = *(const v2f*)(eb + k0 + 16 * QDIM);
                v2f b2 = *(const v2f*)(eb + k0 + 32 * QDIM);
                v2f b3 = *(const v2f*)(eb + k0 + 48 * QDIM);
                // (neg_a, A, neg_b, B, c_mod, C, reuse_a, reuse_b)
                acc0 = __builtin_amdgcn_wmma_f32_16x16x4_f32(false, a, false, b0, (short)0, acc0, false, false);
                acc1 = __builtin_amdgcn_wmma_f32_16x16x4_f32(false, a, false, b1, (short)0, acc1, false, false);
                acc2 = __builtin_amdgcn_wmma_f32_16x16x4_f32(false, a, false, b2, (short)0, acc2, false, false);
                acc3 = __builtin_amdgcn_wmma_f32_16x16x4_f32(false, a, false, b3, (short)0, acc3, false, false);
            }
            #pragma unroll
            for (int v = 0; v < 8; ++v) {
                dst[(mbase + v) * SROW + n0 +  0 + mm] = acc0[v];
                dst[(mbase + v) * SROW + n0 + 16 + mm] = acc1[v];
                dst[(mbase + v) * SROW + n0 + 32 + mm] = acc2[v];
                dst[(mbase + v) * SROW + n0 + 48 + mm] = acc3[v];
            }
        }
        __syncthreads();

        float* tmp = src; src = dst; dst = tmp;   // result of this layer now in src
    }

    // ---------------- 3. parity observable: sum(psi^2 first half) - sum(second half) ----------------
    {
        int h   = lane >> 4;                 // 0: first 256, 1: last 256
        int row = lane & 15;
        const float* p = src + row * SROW + h * (QDIM / 2);
        float partial = 0.0f;
        for (int j = 0; j < QDIM / 2; ++j) {
            float v = p[j];
            partial += v * v;
        }
        partial = h ? -partial : partial;
        partial += __shfl_xor(partial, 16, 32);
        if (lane < 16)
            out[m0 + row] = partial;
    }
}

extern "C" void kernel_launch(void* const* d_in, const int* in_sizes, int n_in,
                              void* d_out, int out_size, void* d_ws, size_t ws_size,
                              hipStream_t stream) {
    const float* x     = (const float*)d_in[0];   // (16,16,4096) f32
    const float* E     = (const float*)d_in[1];   // (512,512)    f32
    const float* theta = (const float*)d_in[2];   // (3,9)        f32
    float* out = (float*)d_out;                   // (16,1,1,4089) f32 flat

    const int total_rows = NBATCH * LOUT;         // 65424 = 4089 * 16, exact
    dim3 grid(total_rows / MROWS);                // 4089 single-wave workgroups
    dim3 block(32);
    qconv_wmma_kernel<<<grid, block, 0, stream>>>(x, E, theta, out);
}